// Attribute_Decoder_81131932221578
// MI455X (gfx1250) — compile-verified
//
#include <hip/hip_runtime.h>

typedef __attribute__((ext_vector_type(2))) float v2f;
typedef __attribute__((ext_vector_type(8))) float v8f;
typedef __attribute__((ext_vector_type(4))) unsigned int u32x4;
typedef __attribute__((ext_vector_type(8))) int i32x8;
typedef __attribute__((ext_vector_type(4))) int i32x4;

// ---------------------------------------------------------------------------
// Degree / normalization kernels (deg and dinv share one buffer)
// ---------------------------------------------------------------------------
__global__ void deg_init_kernel(float* __restrict__ deg, int N) {
    int i = blockIdx.x * blockDim.x + threadIdx.x;
    if (i < N) deg[i] = 1.0f;   // self-loop contributes 1 to every node's degree
}

__global__ void deg_edges_kernel(const long long* __restrict__ cols,
                                 float* __restrict__ deg, long long E) {
    long long e = (long long)blockIdx.x * blockDim.x + threadIdx.x;
    long long stride = (long long)gridDim.x * blockDim.x;
    for (; e < E; e += stride) {
        unsafeAtomicAdd(&deg[cols[e]], 1.0f);
    }
}

__global__ void make_dinv_kernel(float* __restrict__ d, int N) {
    int i = blockIdx.x * blockDim.x + threadIdx.x;
    if (i < N) d[i] = rsqrtf(d[i]);   // deg >= 1 always (self-loop)
}

// ---------------------------------------------------------------------------
// fp32 WMMA GEMM: C[M x Nout] = A[M x K] * W[K x Nout], row-major.
// Block = 8 waves. Each wave computes a 16x64 C tile (4 accumulators) so each
// A fragment feeds 4 WMMAs. The shared W slice W[0:K, n0:n0+64] (64 KB) is
// staged into LDS by the Tensor Data Mover: each wave issues one
// tensor_load_to_lds for its 32-row sub-tile, waits on TENSORcnt, barriers.
// Requires: M % 16 == 0, K % 32 == 0 (K <= 256), Nout % 64 == 0.
// ---------------------------------------------------------------------------
__global__ __launch_bounds__(256) void gemm_wmma_f32_kernel(
    const float* __restrict__ A, const float* __restrict__ W,
    float* __restrict__ C, int M, int K, int Nout)
{
    __shared__ float ldsW[256 * 64];     // [k][n] slice of W, 64 KB max

    const int tid  = threadIdx.x;
    const int wave = tid >> 5;
    const int lane = tid & 31;
    const int n0   = blockIdx.y * 64;

    // ---- TDM stage: wave `w` DMAs W[w*rows:(w+1)*rows, n0:n0+64] to LDS ----
    {
        const int rows = K >> 3;                       // rows per wave (32)
        // Low 32 bits of a generic pointer to LDS == wave-relative LDS byte
        // offset (flat->LDS maps by address truncation, ISA 10.2).
        const unsigned lds_off =
            (unsigned)(unsigned long long)(const void*)&ldsW[(size_t)wave * rows * 64];
        const unsigned long long ga =
            (unsigned long long)(const void*)(W + (long long)wave * rows * Nout + n0);

        // D# group 0: count=1 | lds_addr | global_addr | type=2 ("image")
        u32x4 g0;
        g0[0] = 1u;                                    // count=1, user mode
        g0[1] = lds_off;                               // lds_addr[31:0]
        g0[2] = (unsigned)(ga & 0xFFFFFFFFull);        // global_addr[31:0]
        g0[3] = (unsigned)((ga >> 32) & 0x01FFFFFFull) // global_addr[56:32]
                | (2u << 30);                          // type=2

        // D# group 1: data_size=4B, tensor 2D = rows x Nout, tile = rows x 64
        i32x8 g1;
        g1[0] = (int)(2u << 16);                             // data_size=2 (4B)
        g1[1] = (int)(((unsigned)Nout & 0xFFFFu) << 16);     // tensor_dim0[15:0]
        g1[2] = (int)((((unsigned)Nout >> 16) & 0xFFFFu)     // tensor_dim0[31:16]
                | (((unsigned)rows & 0xFFFFu) << 16));       // tensor_dim1[15:0]
        g1[3] = (int)((((unsigned)rows >> 16) & 0xFFFFu)     // tensor_dim1[31:16]
                | (64u << 16));                              // tile_dim0 = 64
        g1[4] = (int)((unsigned)rows & 0xFFFFu);             // tile_dim1 = rows
        g1[5] = (int)Nout;                                   // dim0_stride[31:0]
        g1[6] = 0;                                           // dim0_stride[47:32], dim1_stride lo
        g1[7] = 0;

        i32x4 gz  = {0, 0, 0, 0};                            // groups 2/3 unused (2D)
        i32x8 gz8 = {0, 0, 0, 0, 0, 0, 0, 0};                // descriptor extension unused

        __builtin_amdgcn_tensor_load_to_lds(g0, g1, gz, gz, gz8, /*cpol=*/0);
        __builtin_amdgcn_s_wait_tensorcnt(0);                // own DMA done
    }
    // The TDM wrote LDS behind the compiler's back (descriptor carries the
    // address as an integer). Escape ldsW + memory clobber so the ds_load
    // reads below are not folded away.
    asm volatile("" : : "v"((const void*)ldsW) : "memory");
    __syncthreads();                                         // all slices visible

    const int mtile = blockIdx.x * 8 + wave;
    if (mtile * 16 >= M) return;         // wave-uniform: EXEC stays all-ones

    // A-matrix 16x4 fragment layout (ISA 7.12.2):
    //   lanes 0-15 : row M=lane,    VGPR0/1 = k+0, k+1
    //   lanes 16-31: row M=lane-16, VGPR0/1 = k+2, k+3
    const int mlocal = lane & 15;
    const int koff   = (lane >> 4) << 1;          // 0 or 2
    const float* arow = A + (long long)(mtile * 16 + mlocal) * K;

    v8f acc0 = {0.f,0.f,0.f,0.f,0.f,0.f,0.f,0.f};
    v8f acc1 = acc0, acc2 = acc0, acc3 = acc0;

    for (int k = 0; k < K; k += 4) {
        // A fragment: contiguous float2 from this lane's row (reused 4x)
        v2f a = *(const v2f*)(arow + k + koff);
        const float* w0 = &ldsW[(k + koff) * 64 + mlocal];
        const float* w1 = &ldsW[(k + koff + 1) * 64 + mlocal];
        v2f b0, b1, b2, b3;
        b0.x = w0[0];  b0.y = w1[0];
        b1.x = w0[16]; b1.y = w1[16];
        b2.x = w0[32]; b2.y = w1[32];
        b3.x = w0[48]; b3.y = w1[48];
        acc0 = __builtin_amdgcn_wmma_f32_16x16x4_f32(false, a, false, b0,
                                                     (short)0, acc0, false, false);
        acc1 = __builtin_amdgcn_wmma_f32_16x16x4_f32(false, a, false, b1,
                                                     (short)0, acc1, false, false);
        acc2 = __builtin_amdgcn_wmma_f32_16x16x4_f32(false, a, false, b2,
                                                     (short)0, acc2, false, false);
        acc3 = __builtin_amdgcn_wmma_f32_16x16x4_f32(false, a, false, b3,
                                                     (short)0, acc3, false, false);
    }

    // C/D 16x16 layout: VGPR r -> row M = r (+8 for lanes 16-31), col = lane&15
    const int mbase = mtile * 16 + ((lane >> 4) << 3);
    float* crow = C + (long long)mbase * Nout + n0 + mlocal;
#pragma unroll
    for (int r = 0; r < 8; ++r) {
        float* cr = crow + (long long)r * Nout;
        cr[0]  = acc0[r];
        cr[16] = acc1[r];
        cr[32] = acc2[r];
        cr[48] = acc3[r];
    }
}

// ---------------------------------------------------------------------------
// out[i][j] = b[j] + xw[i][j] * dinv[i]^2   (bias + self-loop term; also
// fully initializes the accumulation buffer)
// ---------------------------------------------------------------------------
__global__ void init_out_kernel(const float* __restrict__ xw,
                                const float* __restrict__ dinv,
                                const float* __restrict__ bias,
                                float* __restrict__ out,
                                long long total, int F)
{
    long long idx = (long long)blockIdx.x * blockDim.x + threadIdx.x;
    long long stride = (long long)gridDim.x * blockDim.x;
    for (; idx < total; idx += stride) {
        long long i = idx / F;
        int j = (int)(idx - i * F);
        float di = dinv[i];
        out[idx] = bias[j] + xw[idx] * di * di;
    }
}

// ---------------------------------------------------------------------------
// Edge scatter: out[col] += xw[row] * dinv[row]*dinv[col]
// G = F/4 lanes cooperate on one edge using float4 gathers + float atomics.
// ---------------------------------------------------------------------------
template <int F>
__global__ __launch_bounds__(256) void scatter_edges_kernel(
    const long long* __restrict__ rows, const long long* __restrict__ cols,
    const float* __restrict__ dinv, const float* __restrict__ xw,
    float* __restrict__ out, long long E)
{
    constexpr int G   = F / 4;      // lanes per edge
    constexpr int EPB = 256 / G;    // edges per block per iteration
    const int l = threadIdx.x % G;
    long long e = (long long)blockIdx.x * EPB + threadIdx.x / G;
    const long long stride = (long long)gridDim.x * EPB;
    for (; e < E; e += stride) {
        long long r = rows[e];
        long long c = cols[e];
        float nrm = dinv[r] * dinv[c];
        float4 v = ((const float4*)(xw + r * F))[l];
        float* o = out + c * F + l * 4;
        unsafeAtomicAdd(o + 0, v.x * nrm);
        unsafeAtomicAdd(o + 1, v.y * nrm);
        unsafeAtomicAdd(o + 2, v.z * nrm);
        unsafeAtomicAdd(o + 3, v.w * nrm);
    }
}

__global__ void relu_kernel(float* __restrict__ h, long long n) {
    long long i = (long long)blockIdx.x * blockDim.x + threadIdx.x;
    long long stride = (long long)gridDim.x * blockDim.x;
    for (; i < n; i += stride) {
        h[i] = fmaxf(h[i], 0.0f);
    }
}

// ---------------------------------------------------------------------------
// Host launch
// ---------------------------------------------------------------------------
extern "C" void kernel_launch(void* const* d_in, const int* in_sizes, int n_in,
                              void* d_out, int out_size, void* d_ws, size_t ws_size,
                              hipStream_t stream) {
    const float*      x   = (const float*)d_in[0];
    const long long*  ei  = (const long long*)d_in[1];   // int64 edge_index [2,E]
    const float*      W1  = (const float*)d_in[2];
    const float*      b1  = (const float*)d_in[3];
    const float*      W2  = (const float*)d_in[4];
    const float*      b2  = (const float*)d_in[5];
    float*            out = (float*)d_out;

    const int       K  = 256;                     // nhid
    const int       N  = in_sizes[0] / K;         // 50000
    const long long E  = in_sizes[1] / 2;         // 800000
    const int       F1 = in_sizes[2] / K;         // 256
    const int       F2 = in_sizes[4] / K;         // 128

    // Workspace layout: [dinv: N floats][xw: N*F1 floats (reused)][h: N*F1 floats]
    char* ws = (char*)d_ws;
    float* dinv = (float*)ws;
    size_t off1 = ((size_t)N * 4 + 255) & ~(size_t)255;
    float* xw = (float*)(ws + off1);
    size_t off2 = off1 + (((size_t)N * F1 * 4 + 255) & ~(size_t)255);
    float* h = (float*)(ws + off2);

    const long long* rows = ei;        // source nodes
    const long long* cols = ei + E;    // target nodes

    // --- shared normalization ---
    deg_init_kernel<<<(N + 255) / 256, 256, 0, stream>>>(dinv, N);
    deg_edges_kernel<<<(int)((E + 255) / 256), 256, 0, stream>>>(cols, dinv, E);
    make_dinv_kernel<<<(N + 255) / 256, 256, 0, stream>>>(dinv, N);

    const int mblocks = (N / 16 + 7) / 8;   // 8 M-tiles (waves) per block

    // --- layer 1: h = relu(Ahat @ (x @ W1) + b1) ---
    gemm_wmma_f32_kernel<<<dim3(mblocks, F1 / 64), 256, 0, stream>>>(
        x, W1, xw, N, K, F1);
    {
        long long total = (long long)N * F1;
        init_out_kernel<<<(int)((total + 255) / 256), 256, 0, stream>>>(
            xw, dinv, b1, h, total, F1);
        scatter_edges_kernel<256><<<(int)((E + 3) / 4), 256, 0, stream>>>(
            rows, cols, dinv, xw, h, E);
        relu_kernel<<<(int)((total + 255) / 256), 256, 0, stream>>>(h, total);
    }

    // --- layer 2: out = Ahat @ (h @ W2) + b2 ---
    gemm_wmma_f32_kernel<<<dim3(mblocks, F2 / 64), 256, 0, stream>>>(
        h, W2, xw, N, K, F2);
    {
        long long total = (long long)N * F2;
        init_out_kernel<<<(int)((total + 255) / 256), 256, 0, stream>>>(
            xw, dinv, b2, out, total, F2);
        scatter_edges_kernel<128><<<(int)((E + 7) / 8), 256, 0, stream>>>(
            rows, cols, dinv, xw, out, E);
    }
}